// MonotoneFeatureTransform_80865644249209
// MI455X (gfx1250) — compile-verified
//
#include <hip/hip_runtime.h>
#include <hip/hip_bf16.h>
#include <math.h>

#define BOUNDV  5.0f
#define MIN_D   1e-3f
#define MIN_BW  1e-3f
#define EPSV    1e-8f
#define BTOT    131072
#define FDIM    256
#define KBINS   16
#define NKNOT   17
#define NB      512                       // partial-reduction blocks
#define PARAM_FLOATS (3 * NKNOT * FDIM)   // 13056 floats = 52224 B (16B multiple)

typedef __attribute__((ext_vector_type(2))) float v2f;
typedef __attribute__((ext_vector_type(8))) float v8f;

// ---------------------------------------------------------------------------
// Kernel 0: per-feature spline parameter tables, stored transposed [knot][F]
// so that LDS access lds[k*256 + f] is bank-conflict-free (256 % 64 == 0).
// ---------------------------------------------------------------------------
__global__ void __launch_bounds__(256) precompute_params(
    const float* __restrict__ uw, const float* __restrict__ uh,
    const float* __restrict__ ud, float* __restrict__ ws) {
  const int f = threadIdx.x;
  float* cw = ws;
  float* ch = ws + NKNOT * FDIM;
  float* dd = ws + 2 * NKNOT * FDIM;

  // --- width knots ---
  {
    float u[KBINS]; float m = -1e30f;
#pragma unroll
    for (int k = 0; k < KBINS; ++k) { u[k] = uw[f * KBINS + k]; m = fmaxf(m, u[k]); }
    float s = 0.f;
#pragma unroll
    for (int k = 0; k < KBINS; ++k) { u[k] = expf(u[k] - m); s += u[k]; }
    const float inv = 1.f / s;
    float c = 0.f;
    cw[f] = -BOUNDV;
#pragma unroll
    for (int k = 0; k < KBINS; ++k) {
      const float wk = MIN_BW + (1.f - MIN_BW * (float)KBINS) * (u[k] * inv);
      c += wk;
      cw[(k + 1) * FDIM + f] = 2.f * BOUNDV * c - BOUNDV;
    }
    cw[KBINS * FDIM + f] = BOUNDV;
  }
  // --- height knots ---
  {
    float u[KBINS]; float m = -1e30f;
#pragma unroll
    for (int k = 0; k < KBINS; ++k) { u[k] = uh[f * KBINS + k]; m = fmaxf(m, u[k]); }
    float s = 0.f;
#pragma unroll
    for (int k = 0; k < KBINS; ++k) { u[k] = expf(u[k] - m); s += u[k]; }
    const float inv = 1.f / s;
    float c = 0.f;
    ch[f] = -BOUNDV;
#pragma unroll
    for (int k = 0; k < KBINS; ++k) {
      const float wk = MIN_BW + (1.f - MIN_BW * (float)KBINS) * (u[k] * inv);
      c += wk;
      ch[(k + 1) * FDIM + f] = 2.f * BOUNDV * c - BOUNDV;
    }
    ch[KBINS * FDIM + f] = BOUNDV;
  }
  // --- knot derivatives (linear-tail padding) ---
  const float cst = logf(expm1f(1.0f - MIN_D));
#pragma unroll
  for (int k = 0; k < NKNOT; ++k) {
    const float v  = (k == 0 || k == KBINS) ? cst : ud[f * (KBINS - 1) + (k - 1)];
    const float sp = (v > 20.f) ? v : log1pf(expf(v));
    dd[k * FDIM + f] = MIN_D + sp;
  }
}

// ---------------------------------------------------------------------------
// Async global -> LDS fill of the 52 KB parameter table (CDNA5 async path).
// ---------------------------------------------------------------------------
__device__ __forceinline__ void load_params_to_lds(const float* __restrict__ gsrc,
                                                   float* lds) {
  const unsigned lds_base = (unsigned)(size_t)(void*)lds;   // low 32 bits = LDS offset
  const int nvec = PARAM_FLOATS / 4;                        // 3264 x b128 chunks
  for (int i = threadIdx.x; i < nvec; i += (int)blockDim.x) {
    unsigned loff = lds_base + (unsigned)i * 16u;
    unsigned long long gaddr = (unsigned long long)(size_t)(const void*)(gsrc + i * 4);
    asm volatile("global_load_async_to_lds_b128 %0, %1, off"
                 :: "v"(loff), "v"(gaddr) : "memory");
  }
  asm volatile("s_wait_asynccnt 0" ::: "memory");
  __syncthreads();
}

// ---------------------------------------------------------------------------
// RQ-spline eval; lds = [cw[17][256] | ch[17][256] | d[17][256]].
// 4-step binary search == searchsorted(cw, xc, 'right') - 1, clipped.
// ---------------------------------------------------------------------------
__device__ __forceinline__ float eval_spline(float xn, const float* lds, int f) {
  const int S = NKNOT * FDIM;
  const bool inside = (xn > -BOUNDV) && (xn < BOUNDV);
  const float xc = fminf(fmaxf(xn, -BOUNDV), BOUNDV);
  int lo = 0;
#pragma unroll
  for (int step = 8; step >= 1; step >>= 1) {
    const int mid = lo + step;                 // max reachable index 15
    lo = (lds[mid * FDIM + f] <= xc) ? mid : lo;
  }
  const int idx = lo;                          // in [0, 15]
  const float icw = lds[idx * FDIM + f];
  const float cw1 = lds[(idx + 1) * FDIM + f];
  const float ich = lds[S + idx * FDIM + f];
  const float ch1 = lds[S + (idx + 1) * FDIM + f];
  const float d0  = lds[2 * S + idx * FDIM + f];
  const float d1  = lds[2 * S + (idx + 1) * FDIM + f];
  const float ibw   = cw1 - icw;
  const float ibh   = ch1 - ich;
  const float delta = ibh / ibw;
  const float theta = (xc - icw) / ibw;
  const float t1m   = theta * (1.f - theta);
  const float numer = ibh * (delta * theta * theta + d0 * t1m);
  const float denom = delta + (d0 + d1 - 2.f * delta) * t1m;
  return inside ? (ich + numer / denom) : xn;
}

// ---------------------------------------------------------------------------
// Pass 1: streaming reduce -> per-block (sum, sumsq) partials per feature.
// Thread t owns feature t: fully coalesced 128 B per-wave row reads.
// ---------------------------------------------------------------------------
__global__ void __launch_bounds__(256) pass1(
    const float* __restrict__ x, const float* __restrict__ wsp,
    const float* __restrict__ shift, const float* __restrict__ scale,
    float* __restrict__ psum, float* __restrict__ psq) {
  __shared__ __align__(16) float lds_p[PARAM_FLOATS];
  load_params_to_lds(wsp, lds_p);
  const int f = threadIdx.x;
  const float sh = shift[f];
  const float inv_sc = 1.f / scale[f];
  float sum = 0.f, sq = 0.f;
#pragma unroll 4
  for (int b = blockIdx.x; b < BTOT; b += NB) {
    const int pb = min(b + 8 * NB, BTOT - 1);              // branchless prefetch row
    __builtin_prefetch(&x[(size_t)pb * FDIM + f], 0, 3);
    const float xv = x[(size_t)b * FDIM + f];
    const float s = eval_spline((xv - sh) * inv_sc, lds_p, f);
    sum += s; sq += s * s;
  }
  psum[blockIdx.x * FDIM + f] = sum;
  psq [blockIdx.x * FDIM + f] = sq;
}

// ---------------------------------------------------------------------------
// Stage 2: reduce 512 partials/feature with V_WMMA_F32_16X16X4_F32.
// A = ones(16x4) -> D[m][n] = sum_k B[k][n] + C[m][n]; 16 waves x 16 features.
// C/D layout: VGPR0 of lane n (n<16) holds (M=0, N=n) -> the feature total.
// ---------------------------------------------------------------------------
__global__ void __launch_bounds__(512) stage2(
    const float* __restrict__ psum, const float* __restrict__ psq,
    float* __restrict__ meanv, float* __restrict__ istdv) {
  const int lane = threadIdx.x & 31;
  const int wv   = threadIdx.x >> 5;          // 0..15: feature group
  const int n    = lane & 15;
  const int rsel = (lane >> 4) << 1;          // lanes 0-15: K rows {0,1}; 16-31: {2,3}
  const int col  = wv * 16 + n;
#if __has_builtin(__builtin_amdgcn_wmma_f32_16x16x4_f32)
  v8f accS = {0.f, 0.f, 0.f, 0.f, 0.f, 0.f, 0.f, 0.f};
  v8f accQ = {0.f, 0.f, 0.f, 0.f, 0.f, 0.f, 0.f, 0.f};
  v2f ones = {1.f, 1.f};
  for (int b0 = 0; b0 < NB; b0 += 4) {
    const int r = b0 + rsel;
    v2f bS = {psum[r * FDIM + col], psum[(r + 1) * FDIM + col]};
    v2f bQ = {psq [r * FDIM + col], psq [(r + 1) * FDIM + col]};
    accS = __builtin_amdgcn_wmma_f32_16x16x4_f32(false, ones, false, bS,
                                                 (short)0, accS, false, false);
    accQ = __builtin_amdgcn_wmma_f32_16x16x4_f32(false, ones, false, bQ,
                                                 (short)0, accQ, false, false);
  }
  const float tS = accS[0];
  const float tQ = accQ[0];
#else
  float tS = 0.f, tQ = 0.f;
  for (int r = 0; r < NB; ++r) { tS += psum[r * FDIM + col]; tQ += psq[r * FDIM + col]; }
#endif
  if (lane < 16) {
    const float mean = tS / (float)BTOT;
    const float var  = tQ / (float)BTOT - mean * mean;
    meanv[col] = mean;
    istdv[col] = rsqrtf(var + EPSV);
  }
}

// ---------------------------------------------------------------------------
// Pass 2: recompute s (x re-read hits 192 MB L2), normalize, write z with a
// non-temporal store so the z stream does not evict x from L2.
// ---------------------------------------------------------------------------
__global__ void __launch_bounds__(256) pass2(
    const float* __restrict__ x, const float* __restrict__ wsp,
    const float* __restrict__ shift, const float* __restrict__ scale,
    const float* __restrict__ meanv, const float* __restrict__ istdv,
    float* __restrict__ z) {
  __shared__ __align__(16) float lds_p[PARAM_FLOATS];
  load_params_to_lds(wsp, lds_p);
  const int f = threadIdx.x;
  const float sh = shift[f];
  const float inv_sc = 1.f / scale[f];
  const float mu = meanv[f];
  const float is = istdv[f];
#pragma unroll 4
  for (int b = blockIdx.x; b < BTOT; b += NB) {
    const int pb = min(b + 8 * NB, BTOT - 1);              // branchless prefetch row
    __builtin_prefetch(&x[(size_t)pb * FDIM + f], 0, 3);
    const float xv = x[(size_t)b * FDIM + f];
    const float s = eval_spline((xv - sh) * inv_sc, lds_p, f);
    __builtin_nontemporal_store((s - mu) * is, &z[(size_t)b * FDIM + f]);
  }
}

// ---------------------------------------------------------------------------
extern "C" void kernel_launch(void* const* d_in, const int* in_sizes, int n_in,
                              void* d_out, int out_size, void* d_ws, size_t ws_size,
                              hipStream_t stream) {
  (void)in_sizes; (void)n_in; (void)out_size; (void)ws_size;
  const float* x     = (const float*)d_in[0];
  const float* uw    = (const float*)d_in[1];
  const float* uh    = (const float*)d_in[2];
  const float* ud    = (const float*)d_in[3];
  const float* shift = (const float*)d_in[4];
  const float* scale = (const float*)d_in[5];
  float* z  = (float*)d_out;
  float* ws = (float*)d_ws;

  float* params = ws;                                   // 13056 floats
  float* psum   = ws + PARAM_FLOATS;                    // NB*256
  float* psq    = psum + NB * FDIM;                     // NB*256
  float* meanv  = psq + NB * FDIM;                      // 256
  float* istdv  = meanv + FDIM;                         // 256

  precompute_params<<<1, 256, 0, stream>>>(uw, uh, ud, params);
  pass1<<<NB, 256, 0, stream>>>(x, params, shift, scale, psum, psq);
  stage2<<<1, 512, 0, stream>>>(psum, psq, meanv, istdv);
  pass2<<<NB, 256, 0, stream>>>(x, params, shift, scale, meanv, istdv, z);
}